// Group_38843684225797
// MI455X (gfx1250) — compile-verified
//
#include <hip/hip_runtime.h>

typedef __attribute__((ext_vector_type(2))) float        v2f;
typedef __attribute__((ext_vector_type(8))) float        v8f;
typedef __attribute__((ext_vector_type(4))) unsigned int v4u;
typedef __attribute__((ext_vector_type(8))) int          v8i;
typedef __attribute__((ext_vector_type(4))) int          v4i;

#define BATCH 16
#define NPTS  16384
#define NGRP  512
#define KNN   32
#define CHUNK 2048
#define NCHUNK (NPTS / CHUNK)     /* 8  */
#define CAND   (NCHUNK * KNN)     /* 256 */

// ---------------------------------------------------------------------------
// TDM 1-D copy: nelem f32 from global gptr -> LDS byte address lds_addr.
// D# per CDNA5 ISA ch.8: group0 = {count=1 | lds_addr | global_addr | type=2},
// group1 = {data_size=4B, tensor_dim0=nelem, tensor_dim1=1, tile_dim0=nelem,
//           tile_dim1/2=0 (1-D), dim0_stride=nelem}. Trailing groups zero
// (<=2D tensor). This toolchain's builtin is the 6-arg clang-23 form.
// Tracked by TENSORcnt. nelem must fit tile_dim0 (16 bits).
// ---------------------------------------------------------------------------
__device__ __forceinline__ void tdm_load_1d(const void* gptr, unsigned lds_addr,
                                            unsigned nelem)
{
    const unsigned long long ga = (unsigned long long)(uintptr_t)gptr;
    v4u g0;
    g0.x = 1u;                                               // count=1, user mode
    g0.y = lds_addr;                                         // bits 63:32
    g0.z = (unsigned)ga;                                     // global_addr[31:0]
    g0.w = (unsigned)((ga >> 32) & 0x1FFFFFFull) | (2u << 30); // addr[56:32] | type=2
    v8i g1;
    g1[0] = (int)(2u << 16);                                 // data_size = 4 bytes
    g1[1] = (int)((nelem & 0xFFFFu) << 16);                  // tensor_dim0[15:0]
    g1[2] = (int)((nelem >> 16) | (1u << 16));               // tensor_dim0[31:16] | tensor_dim1=1
    g1[3] = (int)((nelem & 0xFFFFu) << 16);                  // tile_dim0 = nelem
    g1[4] = 0;                                               // tile_dim1=0, tile_dim2=0 (1-D)
    g1[5] = (int)nelem;                                      // tensor_dim0_stride lo
    g1[6] = 0;
    g1[7] = 0;
    const v4i z4 = {0, 0, 0, 0};
    const v8i z8 = {0, 0, 0, 0, 0, 0, 0, 0};
    __builtin_amdgcn_tensor_load_to_lds(g0, g1, z4, z4, z8, 0);
}

// ---------------------------------------------------------------------------
// Kernel 1: farthest point sampling. One workgroup per batch. TDM stages the
// whole batch xyz (192KB) into LDS once (overlapped with distance init), so
// all 512 sequential FPS passes run out of LDS: dist(64KB)+xyz(192KB) fits
// the 320KB WGP LDS. Centers written straight into d_out's center section.
// ---------------------------------------------------------------------------
__global__ __launch_bounds__(1024)
void fps_kernel(const float* __restrict__ xyz, float* __restrict__ centers)
{
    extern __shared__ float smem[];
    float* dist  = smem;                         // NPTS
    float* redD  = smem + NPTS;                  // 32
    int*   redI  = (int*)(smem + NPTS + 32);     // 32
    float* s_xyz = smem + NPTS + 64;             // NPTS*3
    __shared__ int   s_far;
    __shared__ float s_cx, s_cy, s_cz;

    const int b    = blockIdx.x;
    const int tid  = threadIdx.x;
    const int lane = tid & 31;
    const int wave = tid >> 5;
    const float* px = xyz + (size_t)b * NPTS * 3;

    if (wave == 0)
        tdm_load_1d(px, (unsigned)(uintptr_t)s_xyz, NPTS * 3);

    for (int n = tid; n < NPTS; n += 1024) dist[n] = 1e10f;   // overlaps the DMA
    if (tid == 0) s_far = 0;
    __builtin_amdgcn_s_wait_tensorcnt(0);
    __syncthreads();

    for (int g = 0; g < NGRP; ++g) {
        if (tid == 0) {
            const int f = s_far;
            const float cx = s_xyz[f * 3 + 0];
            const float cy = s_xyz[f * 3 + 1];
            const float cz = s_xyz[f * 3 + 2];
            s_cx = cx; s_cy = cy; s_cz = cz;
            float* co = centers + ((size_t)b * NGRP + g) * 3;
            co[0] = cx; co[1] = cy; co[2] = cz;
        }
        __syncthreads();

        const float cx = s_cx, cy = s_cy, cz = s_cz;
        float bestD = -1.0f;
        int   bestI = 0;
        for (int n = tid; n < NPTS; n += 1024) {
            const float dx = s_xyz[n * 3 + 0] - cx;
            const float dy = s_xyz[n * 3 + 1] - cy;
            const float dz = s_xyz[n * 3 + 2] - cz;
            float d = dx * dx + dy * dy + dz * dz;
            d = fminf(dist[n], d);
            dist[n] = d;
            if (d > bestD) { bestD = d; bestI = n; }
        }
        #pragma unroll
        for (int m = 16; m >= 1; m >>= 1) {            // wave32 argmax, ties->low idx
            const float od = __shfl_xor(bestD, m, 32);
            const int   oi = __shfl_xor(bestI, m, 32);
            if (od > bestD || (od == bestD && oi < bestI)) { bestD = od; bestI = oi; }
        }
        if (lane == 0) { redD[wave] = bestD; redI[wave] = bestI; }
        __syncthreads();
        if (wave == 0) {
            bestD = redD[lane];
            bestI = redI[lane];
            #pragma unroll
            for (int m = 16; m >= 1; m >>= 1) {
                const float od = __shfl_xor(bestD, m, 32);
                const int   oi = __shfl_xor(bestI, m, 32);
                if (od > bestD || (od == bestD && oi < bestI)) { bestD = od; bestI = oi; }
            }
            if (lane == 0) s_far = bestI;
        }
        __syncthreads();
    }
}

// ---------------------------------------------------------------------------
// Kernel 2: KNN + gather. One workgroup per (batch, 16-center tile).
// Per 2048-point chunk: TDM stages the chunk's xyz (24KB) into LDS, the 16
// waves compute 16x16 d2 tiles with V_WMMA_F32_16X16X4_F32
//   A=(cx,cy,cz,1), B=(-2x,-2y,-2z,||x||^2), C=||c||^2  ->  D = d2 (fp32),
// transpose through a 128KB LDS buffer, then each wave (= one center) pulls
// its 64 values/lane into VGPRs ONCE and extracts the chunk top-32 by
// successive lexicographic (d,idx)-min — no LDS rescans, no LDS mutation.
// Final top-32 over 256 candidates the same way; lane k gathers neighbor k.
// ---------------------------------------------------------------------------
__global__ __launch_bounds__(512)
void knn_kernel(const float* __restrict__ xyz, const float* __restrict__ centers,
                float* __restrict__ neigh)
{
    extern __shared__ float smem[];
    float* s_d2    = smem;                               // 16 * CHUNK
    float* s_candD = smem + 16 * CHUNK;                  // 16 * CAND
    int*   s_candI = (int*)(s_candD + 16 * CAND);        // 16 * CAND
    float* s_cx    = (float*)(s_candI + 16 * CAND);      // 16
    float* s_cy    = s_cx + 16;
    float* s_cz    = s_cy + 16;
    float* s_c2    = s_cz + 16;
    float* s_xyz   = s_c2 + 16;                          // CHUNK * 3

    const int b    = blockIdx.y;
    const int ct   = blockIdx.x;
    const int tid  = threadIdx.x;
    const int lane = tid & 31;
    const int wave = tid >> 5;            // 16 waves <-> 16 centers
    const int half = lane >> 4;           // WMMA half-wave split
    const int m16  = lane & 15;

    const float* px = xyz + (size_t)b * NPTS * 3;

    if (tid < 16) {
        const float* c = centers + ((size_t)b * NGRP + ct * 16 + tid) * 3;
        const float x = c[0], y = c[1], z = c[2];
        s_cx[tid] = x; s_cy[tid] = y; s_cz[tid] = z;
        s_c2[tid] = x * x + y * y + z * z;
    }
    __syncthreads();

    // A operand and ||c||^2 accumulator (uniform across waves / tiles)
    v2f a;
    a.x = half ? s_cz[m16] : s_cx[m16];
    a.y = half ? 1.0f      : s_cy[m16];
    v8f cacc;
    #pragma unroll
    for (int v = 0; v < 8; ++v) cacc[v] = s_c2[half * 8 + v];

    for (int chunk = 0; chunk < NCHUNK; ++chunk) {
        const int P0 = chunk * CHUNK;

        // ---- TDM: stage this chunk's points into LDS ----
        if (wave == 0)
            tdm_load_1d(px + (size_t)P0 * 3, (unsigned)(uintptr_t)s_xyz, CHUNK * 3);
        __builtin_amdgcn_s_wait_tensorcnt(0);
        __syncthreads();

        // ---- WMMA phase: 128 tiles, 8 per wave, EXEC all-1s ----
        for (int t = wave; t < CHUNK / 16; t += 16) {
            const int pl = t * 16 + m16;                 // chunk-local point
            const float x = s_xyz[pl * 3 + 0];
            const float y = s_xyz[pl * 3 + 1];
            const float z = s_xyz[pl * 3 + 2];
            v2f bb;
            bb.x = half ? (-2.0f * z) : (-2.0f * x);
            bb.y = half ? (x * x + y * y + z * z) : (-2.0f * y);
            v8f d = __builtin_amdgcn_wmma_f32_16x16x4_f32(
                false, a, false, bb, (short)0, cacc, false, false);
            #pragma unroll
            for (int v = 0; v < 8; ++v)
                s_d2[(half * 8 + v) * CHUNK + t * 16 + m16] = d[v];
        }
        __syncthreads();

        // ---- chunk-local top-32 for center `wave`: values live in VGPRs ----
        const float* row = s_d2 + wave * CHUNK;
        float vals[64];
        #pragma unroll
        for (int j = 0; j < 64; ++j) vals[j] = row[lane + j * 32];

        float lastD = -1e30f;
        int   lastC = -1;
        for (int it = 0; it < KNN; ++it) {
            float bd = 1e30f;
            int   bc = 0x7fffffff;
            #pragma unroll
            for (int j = 0; j < 64; ++j) {
                const float d = vals[j];
                const int   c = lane + j * 32;
                const bool newer  = (d > lastD) || (d == lastD && c > lastC);
                const bool better = (d < bd)    || (d == bd    && c < bc);
                if (newer && better) { bd = d; bc = c; }
            }
            #pragma unroll
            for (int m = 16; m >= 1; m >>= 1) {
                const float od = __shfl_xor(bd, m, 32);
                const int   oc = __shfl_xor(bc, m, 32);
                if (od < bd || (od == bd && oc < bc)) { bd = od; bc = oc; }
            }
            lastD = bd;                                  // identical in all lanes
            lastC = bc;
            if (lane == 0) {
                s_candD[wave * CAND + chunk * KNN + it] = bd;
                s_candI[wave * CAND + chunk * KNN + it] = P0 + bc;
            }
        }
        __syncthreads();
    }

    // ---- final top-32 over 256 candidates, register-resident ----
    const float* cD = s_candD + wave * CAND;
    const int*   cI = s_candI + wave * CAND;
    float fD[8];
    int   fI[8];
    #pragma unroll
    for (int j = 0; j < 8; ++j) { fD[j] = cD[lane + j * 32]; fI[j] = cI[lane + j * 32]; }

    float lastD = -1e30f;
    int   lastI = -1;
    int   myI   = 0;
    for (int it = 0; it < KNN; ++it) {
        float bd = 1e30f;
        int   bi = 0x7fffffff;
        #pragma unroll
        for (int j = 0; j < 8; ++j) {
            const float d = fD[j];
            const int   i = fI[j];
            const bool newer  = (d > lastD) || (d == lastD && i > lastI);
            const bool better = (d < bd)    || (d == bd    && i < bi);
            if (newer && better) { bd = d; bi = i; }
        }
        #pragma unroll
        for (int m = 16; m >= 1; m >>= 1) {
            const float od = __shfl_xor(bd, m, 32);
            const int   oi = __shfl_xor(bi, m, 32);
            if (od < bd || (od == bd && oi < bi)) { bd = od; bi = oi; }
        }
        lastD = bd;
        lastI = bi;
        if (lane == it) myI = bi;                 // lane k keeps k-th nearest
    }

    // ---- gather: neighborhood[b, g, k, :] = xyz[idx] - center ----
    const int gc = ct * 16 + wave;
    const float cx = s_cx[wave], cy = s_cy[wave], cz = s_cz[wave];
    const float x = px[myI * 3 + 0] - cx;
    const float y = px[myI * 3 + 1] - cy;
    const float z = px[myI * 3 + 2] - cz;
    float* out = neigh + (((size_t)b * NGRP + gc) * KNN + lane) * 3;
    out[0] = x; out[1] = y; out[2] = z;
}

// ---------------------------------------------------------------------------
extern "C" void kernel_launch(void* const* d_in, const int* in_sizes, int n_in,
                              void* d_out, int out_size, void* d_ws, size_t ws_size,
                              hipStream_t stream) {
    (void)in_sizes; (void)n_in; (void)out_size; (void)d_ws; (void)ws_size;
    const float* xyz = (const float*)d_in[0];
    float* out     = (float*)d_out;
    float* neigh   = out;                                      // (B, G, K, 3)
    float* centers = out + (size_t)BATCH * NGRP * KNN * 3;     // (B, G, 3)

    const size_t shm1 = (size_t)(NPTS + 64 + NPTS * 3) * sizeof(float);      // ~256KB
    fps_kernel<<<dim3(BATCH), dim3(1024), shm1, stream>>>(xyz, centers);

    const size_t shm2 = (size_t)(16 * CHUNK + 2 * 16 * CAND + 64 + CHUNK * 3)
                        * sizeof(float);                                     // ~184KB
    knn_kernel<<<dim3(NGRP / 16, BATCH), dim3(512), shm2, stream>>>(xyz, centers, neigh);
}